// SequenceProgramDecoder_66821101191346
// MI455X (gfx1250) — compile-verified
//
#include <hip/hip_runtime.h>
#include <hip/hip_bf16.h>
#include <math.h>

typedef __attribute__((ext_vector_type(2))) float v2f;
typedef __attribute__((ext_vector_type(8))) float v8f;

#define B_ 128
#define S_ 2048
#define H_ 512
#define V_ 32000
#define CHUNKS_ 4                 // S split for attention occupancy
#define SCHUNK_ (S_ / CHUNKS_)    // 512 rows per block

// ---------------------------------------------------------------------------
// K0: x[b,h] = embedding[idx[b], h]
// ---------------------------------------------------------------------------
__global__ void gather_embed_kernel(const int* __restrict__ idx,
                                    const float* __restrict__ emb,
                                    float* __restrict__ x) {
  int i = blockIdx.x * blockDim.x + threadIdx.x;
  if (i < B_ * H_) {
    int b = i / H_, h = i % H_;
    x[i] = emb[(size_t)idx[b] * H_ + h];
  }
}

// ---------------------------------------------------------------------------
// WMMA fp32 GEMM: D[m,n] = act( sum_k A[m,k] * W[n,k] + bias[n] )
// One wave32 per 16x16 output tile, K stepped by 4 via V_WMMA_F32_16X16X4_F32.
// A (16x4 f32, 2 VGPRs): lane&15 = M row, lane>>4 selects K pair.
// B (4x16 f32, 2 VGPRs): lane&15 = N col, lane>>4 selects K pair.
// C/D (16x16 f32, 8 VGPRs): n = lane&15, m = vgpr + 8*(lane>>4).
// ---------------------------------------------------------------------------
template <bool TANH>
__global__ __launch_bounds__(256) void wmma_gemm_nt_kernel(
    const float* __restrict__ A, const float* __restrict__ W,
    const float* __restrict__ bias, float* __restrict__ D,
    int N, int K, int lda, int ldw, int ldd) {
  const int wavesPerBlock = blockDim.x >> 5;
  const int tile = blockIdx.x * wavesPerBlock + (threadIdx.x >> 5);
  const int lane = threadIdx.x & 31;
  const int tiles_n = N >> 4;
  const int m0 = (tile / tiles_n) << 4;
  const int n0 = (tile % tiles_n) << 4;
  const int half = lane >> 4;   // which K pair this half-wave supplies
  const int mn = lane & 15;     // M row (for A) / N col (for B)

  const float* arow = A + (size_t)(m0 + mn) * lda + half * 2;
  const float* wrow = W + (size_t)(n0 + mn) * ldw + half * 2;

  v8f acc = {};
#pragma unroll 4
  for (int k = 0; k < K; k += 4) {
    v2f a = *(const v2f*)(arow + k);
    v2f b = *(const v2f*)(wrow + k);
    acc = __builtin_amdgcn_wmma_f32_16x16x4_f32(
        /*neg_a=*/false, a, /*neg_b=*/false, b,
        /*c_mod=*/(short)0, acc, /*reuse_a=*/false, /*reuse_b=*/false);
  }

  const float bb = bias[n0 + mn];
#pragma unroll
  for (int i = 0; i < 8; ++i) {
    float v = acc[i] + bb;
    if (TANH) v = tanhf(v);
    D[(size_t)(m0 + half * 8 + i) * ldd + n0 + mn] = v;
  }
}

// ---------------------------------------------------------------------------
// K2: GRU gate elementwise math -> h_new. Writes h_new to d_out slot 1 and to
// the first half of the per-batch [h_new | context] concat buffer.
// ---------------------------------------------------------------------------
__global__ void gru_cell_kernel(const float* __restrict__ gi,
                                const float* __restrict__ gh,
                                const float* __restrict__ hprev,
                                float* __restrict__ hn_out,
                                float* __restrict__ concat_buf) {
  int i = blockIdx.x * blockDim.x + threadIdx.x;
  if (i >= B_ * H_) return;
  int b = i / H_, hh = i % H_;
  const float* gib = gi + (size_t)b * 3 * H_;
  const float* ghb = gh + (size_t)b * 3 * H_;
  float r = 1.f / (1.f + __expf(-(gib[hh] + ghb[hh])));
  float z = 1.f / (1.f + __expf(-(gib[H_ + hh] + ghb[H_ + hh])));
  float n = tanhf(gib[2 * H_ + hh] + r * ghb[2 * H_ + hh]);
  float hp = hprev[i];
  float hnew = (1.f - z) * n + z * hp;
  hn_out[i] = hnew;
  concat_buf[(size_t)b * (2 * H_) + hh] = hnew;
}

// ---------------------------------------------------------------------------
// K3a: partial single-pass attention (online softmax) over one S-chunk.
// Grid: (CHUNKS_, B_) -> 512 blocks so the 512 MB stream can saturate HBM.
// Each block: 8 waves x 64 rows; per row 4x global_load_b128, lane-strided
// dot, 5-step shfl reduce, online-softmax update of a 512-wide register
// context fragment. Produces per-(b,chunk) partial (m, l, unnormalized ctx)
// and raw energies. encoder_outputs is read exactly once in total.
// ---------------------------------------------------------------------------
__global__ __launch_bounds__(256) void attention_partial_kernel(
    const float* __restrict__ enc, const float* __restrict__ concat_buf,
    float* __restrict__ energ, float* __restrict__ pm,
    float* __restrict__ pl, float* __restrict__ pc) {
  __shared__ float s_m[8];
  __shared__ float s_l[8];
  __shared__ float s_c[8][H_];

  const int chunk = blockIdx.x;
  const int b = blockIdx.y;
  const int tid = threadIdx.x;
  const int wave = tid >> 5;
  const int lane = tid & 31;

  // h_new fragment: this lane owns components p*128 + lane*4 + {0..3}
  const float* hrow = concat_buf + (size_t)b * (2 * H_);
  float4 h4[4], c4[4];
#pragma unroll
  for (int p = 0; p < 4; ++p) {
    h4[p] = *(const float4*)(hrow + p * 128 + lane * 4);
    c4[p] = make_float4(0.f, 0.f, 0.f, 0.f);
  }

  float m = -INFINITY, l = 0.f;
  const float* encb = enc + (size_t)b * S_ * H_;
  const int s_beg = chunk * SCHUNK_;
  const int s_end = s_beg + SCHUNK_;

  for (int s = s_beg + wave; s < s_end; s += 8) {
    const float* row = encb + (size_t)s * H_;
    float4 e4[4];
    float part = 0.f;
#pragma unroll
    for (int p = 0; p < 4; ++p) {
      e4[p] = *(const float4*)(row + p * 128 + lane * 4);
      part += h4[p].x * e4[p].x + h4[p].y * e4[p].y +
              h4[p].z * e4[p].z + h4[p].w * e4[p].w;
    }
#pragma unroll
    for (int off = 16; off; off >>= 1) part += __shfl_xor(part, off, 32);
    if (lane == 0) energ[(size_t)b * S_ + s] = part;

    float mn_ = fmaxf(m, part);
    float scale = __expf(m - mn_);
    float p_ = __expf(part - mn_);
    l = l * scale + p_;
    m = mn_;
#pragma unroll
    for (int p = 0; p < 4; ++p) {
      c4[p].x = c4[p].x * scale + p_ * e4[p].x;
      c4[p].y = c4[p].y * scale + p_ * e4[p].y;
      c4[p].z = c4[p].z * scale + p_ * e4[p].z;
      c4[p].w = c4[p].w * scale + p_ * e4[p].w;
    }
  }

  if (lane == 0) { s_m[wave] = m; s_l[wave] = l; }
#pragma unroll
  for (int p = 0; p < 4; ++p)
    *(float4*)&s_c[wave][p * 128 + lane * 4] = c4[p];
  __syncthreads();

  float Mb = s_m[0];
#pragma unroll
  for (int w = 1; w < 8; ++w) Mb = fmaxf(Mb, s_m[w]);
  float Lb = 0.f;
#pragma unroll
  for (int w = 0; w < 8; ++w) Lb += s_l[w] * __expf(s_m[w] - Mb);

  const int pidx = b * CHUNKS_ + chunk;
  if (tid == 0) { pm[pidx] = Mb; pl[pidx] = Lb; }

  // merged (unnormalized, relative to Mb) context for this chunk
  if (tid < 128) {
#pragma unroll
    for (int j = 0; j < 4; ++j) {
      int cc = tid * 4 + j;
      float acc = 0.f;
#pragma unroll
      for (int w = 0; w < 8; ++w) acc += s_c[w][cc] * __expf(s_m[w] - Mb);
      pc[(size_t)pidx * H_ + cc] = acc;
    }
  }
}

// ---------------------------------------------------------------------------
// K3b: merge CHUNKS_ partials per batch -> normalized context (into concat
// buffer second half) and attention weights (from stored raw energies).
// ---------------------------------------------------------------------------
__global__ __launch_bounds__(256) void attention_merge_kernel(
    const float* __restrict__ energ, const float* __restrict__ pm,
    const float* __restrict__ pl, const float* __restrict__ pc,
    float* __restrict__ concat_buf, float* __restrict__ attn_out) {
  const int b = blockIdx.x;
  const int tid = threadIdx.x;

  float cm[CHUNKS_], cl[CHUNKS_];
#pragma unroll
  for (int c = 0; c < CHUNKS_; ++c) {
    cm[c] = pm[b * CHUNKS_ + c];
    cl[c] = pl[b * CHUNKS_ + c];
  }
  float M = cm[0];
#pragma unroll
  for (int c = 1; c < CHUNKS_; ++c) M = fmaxf(M, cm[c]);
  float L = 0.f;
#pragma unroll
  for (int c = 0; c < CHUNKS_; ++c) L += cl[c] * __expf(cm[c] - M);

  if (tid < 128) {
#pragma unroll
    for (int j = 0; j < 4; ++j) {
      int cc = tid * 4 + j;
      float acc = 0.f;
#pragma unroll
      for (int c = 0; c < CHUNKS_; ++c)
        acc += pc[(size_t)(b * CHUNKS_ + c) * H_ + cc] * __expf(cm[c] - M);
      concat_buf[(size_t)b * (2 * H_) + H_ + cc] = acc / L;
    }
  }

  for (int s = tid; s < S_; s += 256)
    attn_out[(size_t)b * S_ + s] = __expf(energ[(size_t)b * S_ + s] - M) / L;
}

// ---------------------------------------------------------------------------
extern "C" void kernel_launch(void* const* d_in, const int* in_sizes, int n_in,
                              void* d_out, int out_size, void* d_ws, size_t ws_size,
                              hipStream_t stream) {
  const int*   idx   = (const int*)d_in[0];
  const float* hprev = (const float*)d_in[1];   // (B,1,H)
  const float* enc   = (const float*)d_in[2];   // (B,S,H)
  const float* emb   = (const float*)d_in[3];   // (V,H)
  const float* w_ih  = (const float*)d_in[4];   // (3H,H)
  const float* w_hh  = (const float*)d_in[5];   // (3H,H)
  const float* b_ih  = (const float*)d_in[6];
  const float* b_hh  = (const float*)d_in[7];
  const float* Wc    = (const float*)d_in[8];   // (H,2H)
  const float* bc    = (const float*)d_in[9];
  const float* Wo    = (const float*)d_in[10];  // (V,H)
  const float* bo    = (const float*)d_in[11];

  float* out         = (float*)d_out;
  float* out_logits  = out;                                   // (B,V)
  float* out_h       = out + (size_t)B_ * V_;                 // (B,1,H)
  float* out_attn    = out_h + (size_t)B_ * H_;               // (B,S,1)

  float* ws        = (float*)d_ws;
  float* ws_x      = ws;                           // B*H
  float* ws_gi     = ws_x + B_ * H_;               // B*3H
  float* ws_gh     = ws_gi + B_ * 3 * H_;          // B*3H
  float* ws_concat = ws_gh + B_ * 3 * H_;          // B*2H  [h_new | context]
  float* ws_co     = ws_concat + B_ * 2 * H_;      // B*H   concat_out
  float* ws_energ  = ws_co + B_ * H_;              // B*S   raw energies
  float* ws_pm     = ws_energ + (size_t)B_ * S_;   // B*CHUNKS_
  float* ws_pl     = ws_pm + B_ * CHUNKS_;         // B*CHUNKS_
  float* ws_pc     = ws_pl + B_ * CHUNKS_;         // B*CHUNKS_*H partial ctx

  // K0: embedding gather
  gather_embed_kernel<<<(B_ * H_ + 255) / 256, 256, 0, stream>>>(idx, emb, ws_x);

  // K1: GRU input/hidden GEMMs (bias fused). tiles = 8 x 96 = 768 -> 96 blocks
  wmma_gemm_nt_kernel<false><<<96, 256, 0, stream>>>(
      ws_x, w_ih, b_ih, ws_gi, 3 * H_, H_, H_, H_, 3 * H_);
  wmma_gemm_nt_kernel<false><<<96, 256, 0, stream>>>(
      hprev, w_hh, b_hh, ws_gh, 3 * H_, H_, H_, H_, 3 * H_);

  // K2: gate elementwise -> h_new
  gru_cell_kernel<<<(B_ * H_ + 255) / 256, 256, 0, stream>>>(
      ws_gi, ws_gh, hprev, out_h, ws_concat);

  // K3a: partial attention, 4 S-chunks x 128 batches = 512 blocks
  attention_partial_kernel<<<dim3(CHUNKS_, B_), 256, 0, stream>>>(
      enc, ws_concat, ws_energ, ws_pm, ws_pl, ws_pc);

  // K3b: merge partials -> context + attention weights
  attention_merge_kernel<<<B_, 256, 0, stream>>>(
      ws_energ, ws_pm, ws_pl, ws_pc, ws_concat, out_attn);

  // K4: concat GEMM + tanh. tiles = 8 x 32 = 256 -> 32 blocks
  wmma_gemm_nt_kernel<true><<<32, 256, 0, stream>>>(
      ws_concat, Wc, bc, ws_co, H_, 2 * H_, 2 * H_, 2 * H_, H_);

  // K5: output projection. tiles = 8 x 2000 = 16000 -> 2000 blocks
  wmma_gemm_nt_kernel<false><<<2000, 256, 0, stream>>>(
      ws_co, Wo, bo, out_logits, V_, H_, H_, H_, V_);
}